// KVEmbedding_56822417326208
// MI455X (gfx1250) — compile-verified
//
#include <hip/hip_runtime.h>

// KV-embedding gather for MI455X (gfx1250, wave32).
//
// out[b,l,0:64] = table[indices[b,l], 0:64]
//   indices: (4096,200) int32 (harness passes integer inputs as int*)
//   table:   (1'000'000, 64) float32   (256 MB)
//   out:     (4096,200,64)  float32    (209.7 MB)
//
// Pure data movement: ~420 MB of HBM traffic => ~18 us floor @ 23.3 TB/s.
// Strategy:
//   * 16 lanes per row, one float4 (b128) per lane: fully coalesced
//     GLOBAL_LOAD_B128 gather + GLOBAL_STORE_B128 stream.
//   * Output stores use the CDNA5 temporal-hint path (NT): the 210 MB
//     write-once stream must not evict table rows from the 192 MB L2.
//     Table loads stay RT so the ~32% repeated keys re-hit L2.
//   * No WMMA: zero-FLOP op; matrix pipes cannot help a gather.

typedef __attribute__((ext_vector_type(4))) float v4f;

__global__ __launch_bounds__(256) void kv_gather_b128_nt(
    const int* __restrict__ indices,   // 819,200 keys
    const v4f* __restrict__ table4,    // table viewed as rows of 16 x float4
    v4f*       __restrict__ out4,      // output viewed as float4 stream
    int n4)                            // total float4 elements = 13,107,200
{
    int tid = blockIdx.x * 256 + threadIdx.x;
    if (tid >= n4) return;

    int lookup = tid >> 4;     // which (b,l) key this lane serves
    int chunk  = tid & 15;     // which 16B slice of the 256B row

    // 16 lanes of a half-wave read the same 4B key -> single merged fetch.
    int row = indices[lookup];

    // row in [0, 1e6): (row<<4)|chunk fits in 24 bits -> compiler can use
    // saddr + 32-bit voffset form of global_load_b128.
    v4f v = table4[(row << 4) | chunk];

    // Non-temporal store: keep the streamed output out of L2 (CDNA5 TH field),
    // preserving L2 capacity for the 256 MB table.
    __builtin_nontemporal_store(v, &out4[tid]);
}

extern "C" void kernel_launch(void* const* d_in, const int* in_sizes, int n_in,
                              void* d_out, int out_size, void* d_ws, size_t ws_size,
                              hipStream_t stream) {
    const int*   indices = (const int*)d_in[0];      // (4096*200,) int32
    const float* table   = (const float*)d_in[1];    // (1e6*64,) f32
    // d_in[2] is the dummy scalar; its contribution is exactly 0.0 in the
    // reference (dummy.sum() * 0.0), so it is ignored.

    float* out = (float*)d_out;                      // 4096*200*64 f32

    int n4     = out_size >> 2;                      // float4 count
    int blocks = (n4 + 255) / 256;                   // 51,200 blocks

    kv_gather_b128_nt<<<blocks, 256, 0, stream>>>(
        indices, (const v4f*)table, (v4f*)out, n4);
}